// SiftLoss_86517821210741
// MI455X (gfx1250) — compile-verified
//
#include <hip/hip_runtime.h>
#include <hip/hip_bf16.h>

// ---------------------------------------------------------------------------
// SiftLoss: random 3D gather (128ch x 100k points) + squared-diff reduction.
// Memory-bound: ~430MB effective traffic -> ~20us floor @ 23.3 TB/s HBM.
// Main kernel: wave-per-point gather (max MLP, coalesced feature reads).
// Final kernel: guard-free partial reduction via V_WMMA_F32_16X16X4_F32.
// ---------------------------------------------------------------------------

typedef __attribute__((ext_vector_type(2))) float v2f;
typedef __attribute__((ext_vector_type(8))) float v8f;

#define CHN   128
#define DIMD  96
#define DIMH  96
#define DIMW  96
#define CSTRIDE (DIMD * DIMH * DIMW)   // 884736 elements between channels

#define NBLOCKS  2048                  // multiple of 64 -> guard-free WMMA reduce
#define NTHREADS 256                   // 8 wave32s per block

__global__ void __launch_bounds__(NTHREADS)
sift_gather_kernel(const float* __restrict__ image,
                   const int*   __restrict__ points,   // [N,3] int32
                   const float* __restrict__ feature,  // [N,128] f32
                   float* __restrict__ partials,       // [NBLOCKS]
                   int npoints)
{
    const int lane        = threadIdx.x & 31;
    const int waveInBlk   = threadIdx.x >> 5;
    const int wavesPerBlk = NTHREADS >> 5;
    const int gwave  = blockIdx.x * wavesPerBlk + waveInBlk;
    const int nwaves = gridDim.x * wavesPerBlk;

    const float inv200 = 1.0f / 200.0f;
    const unsigned c0  = (unsigned)lane * 4u;   // 4 channels per lane

    float acc = 0.0f;

    #pragma unroll 2
    for (int n = gwave; n < npoints; n += nwaves) {
        // All 32 lanes load the same 12 bytes -> broadcast, no extra traffic.
        const int p0 = points[3 * n + 0];
        const int p1 = points[3 * n + 1];
        const int p2 = points[3 * n + 2];
        const unsigned voff = (unsigned)p0 * (DIMH * DIMW)
                            + (unsigned)p1 * DIMW
                            + (unsigned)p2;

        // 4 independent scattered loads per lane (channels 3.4MB apart) --
        // issued back-to-back so LOADcnt pipelines them.
        const float* __restrict__ img = image + (size_t)c0 * CSTRIDE + voff;
        const float g0 = img[0 * CSTRIDE];
        const float g1 = img[1 * CSTRIDE];
        const float g2 = img[2 * CSTRIDE];
        const float g3 = img[3 * CSTRIDE];

        // Coalesced: lane l reads feature[n][4l..4l+3] -> one 512B wave req.
        const float4 f = *(const float4*)(feature + (size_t)n * CHN + c0);

        const float d0 = g0 - f.x * inv200;
        const float d1 = g1 - f.y * inv200;
        const float d2 = g2 - f.z * inv200;
        const float d3 = g3 - f.w * inv200;
        acc = fmaf(d0, d0, acc);
        acc = fmaf(d1, d1, acc);
        acc = fmaf(d2, d2, acc);
        acc = fmaf(d3, d3, acc);
    }

    // wave32 reduction
    #pragma unroll
    for (int s = 16; s > 0; s >>= 1)
        acc += __shfl_xor(acc, s, 32);

    __shared__ float smem[NTHREADS / 32];
    if (lane == 0) smem[waveInBlk] = acc;
    __syncthreads();

    if (threadIdx.x == 0) {
        float s = 0.0f;
        #pragma unroll
        for (int i = 0; i < NTHREADS / 32; ++i) s += smem[i];
        partials[blockIdx.x] = s;
    }
}

// Reduce NBLOCKS partials to a scalar with the f32 WMMA pipe.
// D = A x B + C with B = all-ones: every element of the A fragment adds
// exactly once into each column of C, so (column sum of final C) == total.
// NBLOCKS is a multiple of 64 -> no bounds checks, straight-line b64 loads,
// EXEC stays all-ones throughout (WMMA requirement).
__global__ void __launch_bounds__(32)
sift_wmma_reduce_kernel(const float* __restrict__ partials,
                        float* __restrict__ out)
{
    const int lane = threadIdx.x;   // single wave32

    v8f c = {0.f, 0.f, 0.f, 0.f, 0.f, 0.f, 0.f, 0.f};
    v2f b;  b[0] = 1.0f;  b[1] = 1.0f;

    #pragma unroll 4
    for (int base = 0; base < NBLOCKS; base += 64) {   // 64 values per WMMA
        const float2 p = *(const float2*)(partials + base + lane * 2);
        v2f a;  a[0] = p.x;  a[1] = p.y;
        c = __builtin_amdgcn_wmma_f32_16x16x4_f32(
                /*neg_a=*/false, a, /*neg_b=*/false, b,
                /*c_mod=*/(short)0, c, /*reuse_a=*/false, /*reuse_b=*/false);
    }

    // Column 0 of the 16x16 C fragment: lane 0 holds M=0..7, lane 16 holds
    // M=8..15 (8 VGPRs each) -> sum VGPRs, then combine lane 0 with lane 16.
    float s = c[0] + c[1] + c[2] + c[3] + c[4] + c[5] + c[6] + c[7];
    s += __shfl_xor(s, 16, 32);
    if (lane == 0) out[0] = s;
}

extern "C" void kernel_launch(void* const* d_in, const int* in_sizes, int n_in,
                              void* d_out, int out_size, void* d_ws, size_t ws_size,
                              hipStream_t stream)
{
    const float* image    = (const float*)d_in[0];
    const int*   points   = (const int*)  d_in[1];   // [N,3]
    const float* feature  = (const float*)d_in[2];   // [N,128]
    float*       out      = (float*)d_out;
    float*       partials = (float*)d_ws;            // NBLOCKS floats

    const int npoints = in_sizes[1] / 3;

    sift_gather_kernel<<<NBLOCKS, NTHREADS, 0, stream>>>(
        image, points, feature, partials, npoints);

    sift_wmma_reduce_kernel<<<1, 32, 0, stream>>>(partials, out);
}